// DifferentiableEdgeWeighting_25323127177730
// MI455X (gfx1250) — compile-verified
//
#include <hip/hip_runtime.h>
#include <hip/hip_bf16.h>
#include <math.h>

typedef __attribute__((ext_vector_type(2))) float v2f;
typedef __attribute__((ext_vector_type(8))) float v8f;

#define D_DIM 64
#define H_DIM 128
#define CPITCH 66   // float2 per LDS column: 132 words -> lane banks 4c..4c+3, conflict-free

__device__ __forceinline__ unsigned ordKey(float f) {
  unsigned u = __float_as_uint(f);
  return (u & 0x80000000u) ? ~u : (u | 0x80000000u);
}
__device__ __forceinline__ float ordVal(unsigned k) {
  unsigned u = (k & 0x80000000u) ? (k ^ 0x80000000u) : ~k;
  return __uint_as_float(u);
}

__global__ void init_seg(unsigned* __restrict__ segMax, float* __restrict__ segSum, int numS) {
  int i = blockIdx.x * blockDim.x + threadIdx.x;
  if (i < numS) { segMax[i] = 0u; segSum[i] = 0.0f; }
}

// One wave32 handles 16 edges. A = [es|ea] rows (16x128 fp32), B = W1 (128x128),
// 8 N-tiles x 32 K-steps of V_WMMA_F32_16X16X4_F32 per wave.
// W1 is staged in LDS K-pair-interleaved, column-major: element (kk, c) at
// float2 index c*CPITCH + kk/2, component kk&1. A lane's B fragment is then one
// aligned ds_load_b64 with a small immediate offset (16*k bytes).
__global__ __launch_bounds__(256)
void edge_mlp_cost(const float* __restrict__ embS,
                   const float* __restrict__ embA,
                   const int*   __restrict__ edgeS,
                   const int*   __restrict__ edgeA,
                   const float* __restrict__ W1,
                   const float* __restrict__ b1,
                   const float* __restrict__ W2,
                   const float* __restrict__ b2,
                   const float* __restrict__ logT,
                   float*       __restrict__ outCost,
                   unsigned*    __restrict__ segMax,
                   int E) {
  __shared__ float sW1[H_DIM * CPITCH * 2];   // 128 cols * 66 float2 = 67.6 KB
  __shared__ float sB1[H_DIM];
  __shared__ float sW2[H_DIM];
  __shared__ float sGate[8][16];

  const int tid = threadIdx.x;
  for (int i = tid; i < H_DIM * H_DIM; i += 256) {
    int kk = i >> 7;   // input row 0..127
    int c  = i & 127;  // output col 0..127
    sW1[(c * CPITCH + (kk >> 1)) * 2 + (kk & 1)] = W1[i];
  }
  if (tid < H_DIM) { sB1[tid] = b1[tid]; sW2[tid] = W2[tid]; }
  __syncthreads();

  const int wave = tid >> 5;
  const int lane = tid & 31;
  const int half = lane >> 4;   // which K-pair this lane owns
  const int col  = lane & 15;
  const int e0   = (blockIdx.x * 8 + wave) * 16;
  if (e0 >= E) return;          // wave-uniform: EXEC stays all-ones for WMMA

  const int  e     = e0 + col;
  const bool valid = (e < E);
  const int  ec    = valid ? e : (E - 1);
  const int  sI    = edgeS[ec];
  const int  aI    = edgeA[ec];
  const float* es = embS + (size_t)sI * D_DIM;
  const float* ea = embA + (size_t)aI * D_DIM;

  // A fragments: a[k] = cat[edge][4k + 2*half .. +1]; k<16 from es, k>=16 from ea.
  v2f a[32];
#pragma unroll
  for (int k = 0; k < 16; ++k) {
    a[k]      = *(const v2f*)(es + 4 * k + 2 * half);
    a[k + 16] = *(const v2f*)(ea + 4 * k + 2 * half);
  }
  // Same positions of es/ea land in the same lane -> local ||es-ea||^2 partials.
  float ss = 0.0f;
#pragma unroll
  for (int k = 0; k < 16; ++k) {
    float dx = a[k][0] - a[k + 16][0];
    float dy = a[k][1] - a[k + 16][1];
    ss += dx * dx + dy * dy;
  }
  ss += __shfl_xor(ss, 16);     // both halves now hold full sum for edge `col`

  float gacc[8] = {0, 0, 0, 0, 0, 0, 0, 0};
#pragma unroll
  for (int n = 0; n < 8; ++n) {
    // per-lane B column base: all 32 K fragments are at +16*k bytes from here
    const float* bbase = sW1 + ((n * 16 + col) * CPITCH + half) * 2;
    v8f c0 = {};
    v8f c1 = {};
#pragma unroll
    for (int k = 0; k < 32; k += 2) {
      v2f b0 = *(const v2f*)(bbase + 4 * k);
      v2f b1v = *(const v2f*)(bbase + 4 * (k + 1));
      c0 = __builtin_amdgcn_wmma_f32_16x16x4_f32(false, a[k], false, b0,
                                                 (short)0, c0, false, false);
      c1 = __builtin_amdgcn_wmma_f32_16x16x4_f32(false, a[k + 1], false, b1v,
                                                 (short)0, c1, false, false);
    }
    // fused bias + ReLU + W2 dot on the accumulator fragment
    const float bn  = sB1[n * 16 + col];
    const float w2n = sW2[n * 16 + col];
#pragma unroll
    for (int r = 0; r < 8; ++r) {
      float h = c0[r] + c1[r] + bn;
      h = h > 0.0f ? h : 0.0f;
      gacc[r] += h * w2n;
    }
  }

  // Reduce over N (16 lanes per half): VGPR r holds edge r (lanes 0-15) / r+8 (16-31).
#pragma unroll
  for (int r = 0; r < 8; ++r) {
    float g = gacc[r];
    g += __shfl_xor(g, 1);
    g += __shfl_xor(g, 2);
    g += __shfl_xor(g, 4);
    g += __shfl_xor(g, 8);
    if (lane == 0)  sGate[wave][r]     = g;
    if (lane == 16) sGate[wave][r + 8] = g;
  }
  // same-wave DS ops are processed in order: read-after-write is safe here

  const float b2v  = b2[0];
  const float invT = __expf(-logT[0]);
  if (lane < 16 && valid) {
    float logit = sGate[wave][lane] + b2v;
    float gate  = 1.0f / (1.0f + __expf(-logit));
    float cost  = sqrtf(ss) * gate;
    outCost[e]  = cost;
    atomicMax(&segMax[sI], ordKey(-cost * invT));
  }
}

__global__ void seg_expsum(const float* __restrict__ outCost,
                           const int* __restrict__ edgeS,
                           const unsigned* __restrict__ segMax,
                           float* __restrict__ segSum,
                           const float* __restrict__ logT, int E) {
  int e = blockIdx.x * blockDim.x + threadIdx.x;
  if (e >= E) return;
  float invT = __expf(-logT[0]);
  int s = edgeS[e];
  float val = -outCost[e] * invT;
  float m = ordVal(segMax[s]);
  atomicAdd(&segSum[s], __expf(val - m));
}

__global__ void seg_norm(const float* __restrict__ outCost,
                         const int* __restrict__ edgeS,
                         const unsigned* __restrict__ segMax,
                         const float* __restrict__ segSum,
                         const float* __restrict__ logT,
                         float* __restrict__ outW, int E) {
  int e = blockIdx.x * blockDim.x + threadIdx.x;
  if (e >= E) return;
  float invT = __expf(-logT[0]);
  int s = edgeS[e];
  float val = -outCost[e] * invT;
  float m = ordVal(segMax[s]);
  outW[e] = __expf(val - m) / segSum[s];
}

extern "C" void kernel_launch(void* const* d_in, const int* in_sizes, int n_in,
                              void* d_out, int out_size, void* d_ws, size_t ws_size,
                              hipStream_t stream) {
  const float* embS = (const float*)d_in[0];
  const float* embA = (const float*)d_in[1];
  const int*   eidx = (const int*)d_in[2];
  const float* W1   = (const float*)d_in[3];
  const float* b1   = (const float*)d_in[4];
  const float* W2   = (const float*)d_in[5];
  const float* b2   = (const float*)d_in[6];
  const float* logT = (const float*)d_in[7];

  const int numS = in_sizes[0] / D_DIM;
  const int E    = in_sizes[2] / 2;
  const int* edgeS = eidx;
  const int* edgeA = eidx + E;

  float* outW    = (float*)d_out;       // edge_weights
  float* outCost = (float*)d_out + E;   // edge_costs

  unsigned* segMax = (unsigned*)d_ws;
  float*    segSum = (float*)((unsigned*)d_ws + numS);

  init_seg<<<(numS + 255) / 256, 256, 0, stream>>>(segMax, segSum, numS);

  const int tiles = (E + 15) / 16;       // 16 edges per wave, 8 waves per block
  edge_mlp_cost<<<(tiles + 7) / 8, 256, 0, stream>>>(
      embS, embA, edgeS, edgeA, W1, b1, W2, b2, logT, outCost, segMax, E);

  seg_expsum<<<(E + 255) / 256, 256, 0, stream>>>(outCost, edgeS, segMax, segSum, logT, E);
  seg_norm<<<(E + 255) / 256, 256, 0, stream>>>(outCost, edgeS, segMax, segSum, logT, outW, E);
}